// LCGP_41609643163884
// MI455X (gfx1250) — compile-verified
//
#include <hip/hip_runtime.h>

#define QQ 8
#define DD 8
#define PP 64
#define NN 4096      // n (training points) == K == R of the big GEMM
#define MM 2048      // n0 (test points)   == M
#define KT 32        // K-tile staged in LDS per iteration
#define AS_STRIDE 34    // 32 k dwords + 2 pad dwords (TDM pad: interval=4 -> 32dw, amount=1 -> 2dw)
#define BS_STRIDE 136   // 128 r dwords + 8 pad dwords (TDM pad: interval=6 -> 128dw, amount=7 -> 8dw)
#define RTILES 256      // total 16-wide r-tiles (NN/16)

typedef __attribute__((ext_vector_type(2))) float v2f;
typedef __attribute__((ext_vector_type(8))) float v8f;
typedef __attribute__((ext_vector_type(4))) unsigned int u32x4;
typedef __attribute__((ext_vector_type(4))) int i32x4;
typedef __attribute__((ext_vector_type(8))) int i32x8;

// TDM availability check must apply to the DEVICE pass only: the builtins are
// target builtins, invisible to hipcc's host pass (which still parses kernel
// bodies, so a host-parseable fallback is kept below).
#if defined(__HIP_DEVICE_COMPILE__)
#if !__has_builtin(__builtin_amdgcn_tensor_load_to_lds) || \
    !__has_builtin(__builtin_amdgcn_s_wait_tensorcnt)
#error "CDNA5 TDM builtins not available in device pass"
#endif
#define USE_TDM 1
#else
#define USE_TDM 0   // host pass: parse-only fallback, never codegens device code
#endif

#if USE_TDM
// LDS byte offset of a __shared__ object (addrspacecast generic->as3, then ptrtoint)
__device__ __forceinline__ unsigned lds_off(const void* p)
{
    return (unsigned)(unsigned long long)
        (const __attribute__((address_space(3))) void*)p;
}

// Issue one 2D TDM tile load: tileD1 rows of tileD0 contiguous f32 elements,
// row stride strideD0 elements in global memory, LDS rows padded by the D#
// pad feature (padInterval/padAmount per ISA Group1 encoding).
__device__ __forceinline__ void tdm_load_2d(
    unsigned ldsByteOff, const float* g,
    unsigned tileD0, unsigned tileD1,
    unsigned tensorD0, unsigned tensorD1, unsigned strideD0,
    unsigned padInterval, unsigned padAmount)
{
    unsigned long long ga = (unsigned long long)g;
    u32x4 g0;
    g0[0] = 1u;                                              // count=1, load, user
    g0[1] = ldsByteOff;                                      // lds_addr
    g0[2] = (unsigned)ga;                                    // global_addr[31:0]
    g0[3] = (unsigned)((ga >> 32) & 0x01ffffffu) | (2u << 30); // addr[56:32] | type=2
    i32x8 g1;
    g1[0] = (int)((2u << 16) |                               // data_size = 4B
                  (1u << 20) |                               // pad_enable
                  (padInterval << 22) | (padAmount << 25));  // pad fields
    g1[1] = (int)((tensorD0 & 0xffffu) << 16);               // abar=0 | td0 lo16
    g1[2] = (int)(((tensorD0 >> 16) & 0xffffu) | ((tensorD1 & 0xffffu) << 16));
    g1[3] = (int)(((tensorD1 >> 16) & 0xffffu) | (tileD0 << 16));
    g1[4] = (int)(tileD1 & 0xffffu);                         // tile_dim2 = 0 (2D)
    g1[5] = (int)strideD0;                                   // dim0 stride lo32
    g1[6] = 0;                                               // stride0 hi | stride1 lo
    g1[7] = 0;
    i32x4 z4 = {0, 0, 0, 0};
#if defined(__clang_major__) && __clang_major__ >= 23
    i32x8 z8 = {0, 0, 0, 0, 0, 0, 0, 0};
    __builtin_amdgcn_tensor_load_to_lds(g0, g1, z4, z4, z8, 0);
#else
    __builtin_amdgcn_tensor_load_to_lds(g0, g1, z4, z4, 0);
#endif
}
#endif // USE_TDM

// ---------------------------------------------------------------------------
// Kernel 1: C0[q][m][n] = exp(lLmb0[q] - ||a-b||^2), ghat[q][m] = C0 . CinvM
// ---------------------------------------------------------------------------
__global__ __launch_bounds__(256) void k_cov_ghat(
    const float* __restrict__ x0, const float* __restrict__ x,
    const float* __restrict__ xmin, const float* __restrict__ xmax,
    const float* __restrict__ lLmb, const float* __restrict__ lLmb0,
    const float* __restrict__ CinvM,
    float* __restrict__ C0, float* __restrict__ ghat)
{
    const int q   = blockIdx.x / MM;
    const int m   = blockIdx.x % MM;
    const int tid = threadIdx.x;

    float invl[DD], a[DD];
#pragma unroll
    for (int d = 0; d < DD; ++d) {
        invl[d] = expf(-0.5f * lLmb[q * DD + d]);
        float x0s = (x0[m * DD + d] - xmin[d]) / (xmax[d] - xmin[d]);
        a[d] = x0s * invl[d];
    }
    const float lL0 = lLmb0[q];

    float gacc = 0.0f;
    for (int n = tid; n < NN; n += 256) {
        float sq = 0.0f;
#pragma unroll
        for (int d = 0; d < DD; ++d) {
            float b = x[n * DD + d] * invl[d];
            float diff = a[d] - b;
            sq += diff * diff;
        }
        float c = expf(lL0 - sq);
        C0[(size_t)(q * MM + m) * NN + n] = c;
        gacc += c * CinvM[q * NN + n];
    }

    __shared__ float red[256];
    red[tid] = gacc;
    __syncthreads();
    for (int s = 128; s > 0; s >>= 1) {
        if (tid < s) red[tid] += red[tid + s];
        __syncthreads();
    }
    if (tid == 0) ghat[blockIdx.x] = red[0];
}

// ---------------------------------------------------------------------------
// Kernel 2: fused GEMM-square-reduce, V_WMMA_F32_16X16X4_F32, TDM-staged LDS
// with double buffering. Block = (r_blk 128 cols, m_blk 64 rows, q); 8 waves;
// wave w owns r-tile w x four 16-row m-tiles. Wave 0 DMAs the A tile, wave 1
// the B tile (TENSORcnt is per-wave and in-order, so each waits <=1 after
// prefetching its next tile).
// ---------------------------------------------------------------------------
__global__ __launch_bounds__(256) void k_gemm_sq(
    const float* __restrict__ C0, const float* __restrict__ Th,
    float* __restrict__ part)
{
    const int r_blk = blockIdx.x;           // 0..31
    const int m_blk = blockIdx.y;           // 0..31
    const int q     = blockIdx.z;           // 0..7
    const int tid   = threadIdx.x;
    const int m_base = m_blk * 64;
    const int r_base = r_blk * 128;

    __shared__ __align__(16) float As[2][64 * AS_STRIDE];  // A[m][k], padded rows
    __shared__ __align__(16) float Bs[2][KT * BS_STRIDE];  // B[k][r], padded rows

    const int wv   = tid >> 5;
    const int lane = tid & 31;
    const int half = lane >> 4;
    const int l16  = lane & 15;

    v8f acc[4];
#pragma unroll
    for (int mt = 0; mt < 4; ++mt) acc[mt] = 0.0f;

    const float* Abase = C0 + (size_t)(q * MM + m_base) * NN;   // [64 rows x NN]
    const float* Bbase = Th + (size_t)q * NN * NN + r_base;     // [NN rows x 128]

#if USE_TDM
    unsigned aOff[2] = { lds_off(&As[0][0]), lds_off(&As[1][0]) };
    unsigned bOff[2] = { lds_off(&Bs[0][0]), lds_off(&Bs[1][0]) };
    if (wv == 0) {
        tdm_load_2d(aOff[0], Abase, KT, 64, NN, MM, NN, 4, 1);
    } else if (wv == 1) {
        tdm_load_2d(bOff[0], Bbase, 128, KT, NN, NN, NN, 6, 7);
    }
#endif

    int cur = 0;
    for (int kt = 0; kt < NN; kt += KT) {
#if USE_TDM
        if (kt + KT < NN) {   // prefetch next tile into the other buffer
            if (wv == 0) {
                tdm_load_2d(aOff[cur ^ 1], Abase + (kt + KT), KT, 64, NN, MM, NN, 4, 1);
                __builtin_amdgcn_s_wait_tensorcnt(1);  // in-order: current A landed
            } else if (wv == 1) {
                tdm_load_2d(bOff[cur ^ 1], Bbase + (size_t)(kt + KT) * NN,
                            128, KT, NN, NN, NN, 6, 7);
                __builtin_amdgcn_s_wait_tensorcnt(1);  // in-order: current B landed
            }
        } else if (wv <= 1) {
            __builtin_amdgcn_s_wait_tensorcnt(0);
        }
        __syncthreads();
#else
        {   // host-pass / non-TDM fallback staging (parse-valid everywhere)
            const int a_m = tid >> 2;            // 0..63
            const int a_k = (tid & 3) << 3;      // 0,8,16,24
            const float* ag = Abase + (size_t)a_m * NN + kt + a_k;
            float4 p0 = *(const float4*)ag;
            float4 p1 = *(const float4*)(ag + 4);
            float* ad = &As[cur][a_m * AS_STRIDE + a_k];
            ad[0] = p0.x; ad[1] = p0.y; ad[2] = p0.z; ad[3] = p0.w;
            ad[4] = p1.x; ad[5] = p1.y; ad[6] = p1.z; ad[7] = p1.w;
            const int b_k = tid >> 3;            // 0..31
            const int b_r = (tid & 7) << 4;      // 0..112
            const float4* bsrc = (const float4*)(Bbase + (size_t)(kt + b_k) * NN + b_r);
            float4 q0 = bsrc[0], q1 = bsrc[1], q2 = bsrc[2], q3 = bsrc[3];
            float4* bdst = (float4*)&Bs[cur][b_k * BS_STRIDE + b_r];
            bdst[0] = q0; bdst[1] = q1; bdst[2] = q2; bdst[3] = q3;
        }
        __syncthreads();
#endif

#pragma unroll
        for (int kk = 0; kk < KT; kk += 4) {
            const int bk = kk + half * 2;   // lanes<16: K={kk,kk+1}; lanes>=16: K={kk+2,kk+3}
            v2f bf;
            bf.x = Bs[cur][(bk + 0) * BS_STRIDE + wv * 16 + l16];
            bf.y = Bs[cur][(bk + 1) * BS_STRIDE + wv * 16 + l16];
#pragma unroll
            for (int mt = 0; mt < 4; ++mt) {
                const float2 a2 = *(const float2*)&As[cur][(mt * 16 + l16) * AS_STRIDE + bk];
                v2f af; af.x = a2.x; af.y = a2.y;
                acc[mt] = __builtin_amdgcn_wmma_f32_16x16x4_f32(
                    false, af, false, bf, (short)0, acc[mt], false, false);
            }
        }
        __syncthreads();
        cur ^= 1;
    }

    // Epilogue: D layout: VGPR v, lanes 0-15 = row v, lanes 16-31 = row v+8,
    // column = l16 of this wave's r-tile. Sum of squares over the 16 columns.
    const int rt = r_blk * 8 + wv;   // global r-tile id 0..255
#pragma unroll
    for (int mt = 0; mt < 4; ++mt) {
#pragma unroll
        for (int v = 0; v < 8; ++v) {
            float s = acc[mt][v] * acc[mt][v];
            s += __shfl_xor(s, 8, 32);   // masks <16: stays within each 16-lane half
            s += __shfl_xor(s, 4, 32);
            s += __shfl_xor(s, 2, 32);
            s += __shfl_xor(s, 1, 32);
            if (l16 == 0) {
                const int row = m_base + mt * 16 + half * 8 + v;
                part[(size_t)(q * RTILES + rt) * MM + row] = s;
            }
        }
    }
}

// ---------------------------------------------------------------------------
// Kernel 3: gvar[q][m] = c00[q] - sum_rt part[q][rt][m]   (fixed order)
// ---------------------------------------------------------------------------
__global__ __launch_bounds__(256) void k_gvar(
    const float* __restrict__ part, const float* __restrict__ lLmb0,
    const float* __restrict__ lnug, float* __restrict__ gvar)
{
    const int idx = blockIdx.x * 256 + threadIdx.x;
    const int q = idx / MM;
    const int m = idx % MM;
    float s = 0.0f;
    for (int rt = 0; rt < RTILES; ++rt)
        s += part[(size_t)(q * RTILES + rt) * MM + m];
    const float c00 = expf(lLmb0[q]) * (1.0f + expf(lnug[q]));
    gvar[idx] = c00 - s;
}

// ---------------------------------------------------------------------------
// Kernel 4: outputs. ypred | ypredvar | yconfvar, each [P, MM], concatenated.
// ---------------------------------------------------------------------------
__global__ __launch_bounds__(256) void k_final(
    const float* __restrict__ ghat, const float* __restrict__ gvar,
    const float* __restrict__ lsig, const float* __restrict__ phi,
    const float* __restrict__ ystd, const float* __restrict__ ymean,
    float* __restrict__ out)
{
    const int idx = blockIdx.x * 256 + threadIdx.x;
    const int p = idx / MM;
    const int m = idx % MM;
    const float sig = expf(lsig[p]);
    const float sqs = sqrtf(sig);
    float pm = 0.0f, cv = 0.0f;
#pragma unroll
    for (int q = 0; q < QQ; ++q) {
        float psi = phi[p * QQ + q] * sqs;
        pm += psi * ghat[q * MM + m];
        cv += psi * psi * gvar[q * MM + m];
    }
    const float ys = ystd[p];
    const float ym = ymean[p];
    const float ys2 = ys * ys;
    out[(size_t)p * MM + m]               = pm * ys + ym;       // ypred
    out[(size_t)PP * MM + p * MM + m]     = (cv + sig) * ys2;   // ypredvar
    out[(size_t)2 * PP * MM + p * MM + m] = cv * ys2;           // yconfvar
}

// ---------------------------------------------------------------------------
extern "C" void kernel_launch(void* const* d_in, const int* in_sizes, int n_in,
                              void* d_out, int out_size, void* d_ws, size_t ws_size,
                              hipStream_t stream)
{
    (void)in_sizes; (void)n_in; (void)out_size; (void)ws_size;
    const float* x0    = (const float*)d_in[0];
    const float* x     = (const float*)d_in[1];
    const float* xmin  = (const float*)d_in[2];
    const float* xmax  = (const float*)d_in[3];
    const float* lLmb  = (const float*)d_in[4];
    const float* lLmb0 = (const float*)d_in[5];
    const float* lnug  = (const float*)d_in[6];
    const float* lsig  = (const float*)d_in[7];
    const float* phi   = (const float*)d_in[8];
    const float* CinvM = (const float*)d_in[9];
    const float* Th    = (const float*)d_in[10];
    const float* ystd  = (const float*)d_in[11];
    const float* ymean = (const float*)d_in[12];
    float* out = (float*)d_out;

    // workspace layout (floats): C0 | ghat | part | gvar  (~273 MB total)
    float* C0   = (float*)d_ws;
    float* ghat = C0   + (size_t)QQ * MM * NN;
    float* part = ghat + (size_t)QQ * MM;
    float* gvar = part + (size_t)QQ * RTILES * MM;

    k_cov_ghat<<<QQ * MM, 256, 0, stream>>>(x0, x, xmin, xmax, lLmb, lLmb0,
                                            CinvM, C0, ghat);
    k_gemm_sq<<<dim3(NN / 128, MM / 64, QQ), 256, 0, stream>>>(C0, Th, part);
    k_gvar<<<(QQ * MM) / 256, 256, 0, stream>>>(part, lLmb0, lnug, gvar);
    k_final<<<(PP * MM) / 256, 256, 0, stream>>>(ghat, gvar, lsig, phi,
                                                 ystd, ymean, out);
}